// BaseGP_82987358093858
// MI455X (gfx1250) — compile-verified
//
#include <hip/hip_runtime.h>
#include <math.h>

#define TPTS 8192
#define NB   128
#define NBLK (TPTS / NB)
#define LDA  TPTS
#define LDT  132   // padded LDS row stride (floats) == 128 + 4 (TDM pad: interval 128 dw, amount 4 dw)

typedef float v2f __attribute__((ext_vector_type(2)));
typedef float v8f __attribute__((ext_vector_type(8)));
typedef unsigned int v4u __attribute__((ext_vector_type(4)));
typedef int v8i __attribute__((ext_vector_type(8)));
typedef int v4i __attribute__((ext_vector_type(4)));

#if defined(__has_builtin)
#if __has_builtin(__builtin_amdgcn_tensor_load_to_lds) && __has_builtin(__builtin_amdgcn_s_wait_tensorcnt)
#define HAVE_TDM 1
#endif
#endif
#ifndef HAVE_TDM
#define HAVE_TDM 0
#endif

#if HAVE_TDM
// ---------------------------------------------------------------------------
// TDM: DMA one 128x128 fp32 tile (row stride LDA) from global into LDS with
// 4-DWORD padding after every 128 DWORDs -> LDS row stride LDT=132 floats.
// D# per CDNA5 ISA ch.8: group0 {count=1, lds_addr, global_addr, type=2},
// group1 {data_size=2(4B), pad_enable, pad_interval=6(128dw), pad_amount=3(4dw),
//         tensor_dim0/1=128, tile_dim0/1=128, tensor_dim0_stride=LDA}.
// This toolchain's builtin takes the 6-arg form:
//   (uint32x4 g0, int32x8 g1, int32x4 g2, int32x4 g3, int32x8 g4, i32 cpol)
// ---------------------------------------------------------------------------
__device__ __forceinline__ void tdm_load_tile(const float* gsrc, unsigned lds_off) {
    unsigned long long ga = (unsigned long long)(size_t)gsrc;
    v4u g0;
    g0[0] = 1u;                                              // count=1, user mode
    g0[1] = lds_off;                                         // lds_addr (bytes)
    g0[2] = (unsigned)(ga & 0xFFFFFFFFull);                  // global_addr[31:0]
    g0[3] = (unsigned)((ga >> 32) & 0x1FFFFFFull)            // global_addr[56:32]
          | (2u << 30);                                      // type=2 ("image")
    v8i g1;
    g1[0] = (int)((2u << 16)      // data_size = 4 bytes
                | (1u << 20)      // pad_enable
                | (6u << 22)      // pad_interval: 128 DWORDs
                | (3u << 25));    // pad_amount: 4 DWORDs
    g1[1] = (int)(128u << 16);    // [47:32] atomic_barrier=0, [63:48] tensor_dim0.lo=128
    g1[2] = (int)(128u << 16);    // [79:64] tensor_dim0.hi=0, [95:80] tensor_dim1.lo=128
    g1[3] = (int)(128u << 16);    // [111:96] tensor_dim1.hi=0, [127:112] tile_dim0=128
    g1[4] = (int)128u;            // [143:128] tile_dim1=128, [159:144] tile_dim2=0
    g1[5] = (int)(unsigned)LDA;   // tensor_dim0_stride[31:0] = 8192 elements
    g1[6] = 0;                    // stride hi / tensor_dim1_stride.lo
    g1[7] = 0;
    v4i gz4 = {0, 0, 0, 0};
    v8i gz8 = {0, 0, 0, 0, 0, 0, 0, 0};
    __builtin_amdgcn_tensor_load_to_lds(g0, g1, gz4, gz4, gz8, 0);
}
#endif

// ---------------------------------------------------------------------------
// 1) Build A = variance * exp(-(i-j)^2 / (2 l^2)) + sigma^2 * I   (row-major)
// ---------------------------------------------------------------------------
__global__ void build_A(float* __restrict__ A,
                        const float* __restrict__ sigma_sq,
                        const float* __restrict__ lengthscale,
                        const float* __restrict__ variance) {
    const int j = blockIdx.x * blockDim.x + threadIdx.x;   // column 0..8191
    const int i = blockIdx.y;                              // row    0..8191
    const float l = lengthscale[0];
    const float inv2l2 = 1.0f / (2.0f * l * l);
    const float d = (float)(i - j);
    float v = variance[0] * __expf(-(d * d) * inv2l2);
    if (i == j) v += sigma_sq[0];
    A[(size_t)i * LDA + j] = v;
}

// ---------------------------------------------------------------------------
// 2a) Factor the 128x128 diagonal block in LDS (unblocked right-looking).
// ---------------------------------------------------------------------------
__global__ void panel_factor(float* __restrict__ A, int k) {
    extern __shared__ float sm[];
    float* s = sm;                      // [NB][LDT]
    const int tid = threadIdx.x;        // 128 threads
    float* base = A + (size_t)(k * NB) * LDA + (size_t)k * NB;

    for (int idx = tid; idx < NB * (NB / 4); idx += NB) {
        int r = idx >> 5, c4 = (idx & 31) << 2;
        *(float4*)(s + r * LDT + c4) = *(const float4*)(base + (size_t)r * LDA + c4);
    }
    __syncthreads();

    for (int j = 0; j < NB; ++j) {
        if (tid == j) s[j * LDT + j] = sqrtf(s[j * LDT + j]);
        __syncthreads();
        if (tid > j) s[tid * LDT + j] /= s[j * LDT + j];
        __syncthreads();
        if (tid > j) {
            const float ljt = s[tid * LDT + j];
            for (int c = j + 1; c <= tid; ++c)
                s[tid * LDT + c] -= ljt * s[c * LDT + j];
        }
        __syncthreads();
    }

    for (int idx = tid; idx < NB * (NB / 4); idx += NB) {
        int r = idx >> 5, c4 = (idx & 31) << 2;
        *(float4*)(base + (size_t)r * LDA + c4) = *(float4*)(s + r * LDT + c4);
    }
}

// ---------------------------------------------------------------------------
// 2b) Panel TRSM: for row-block i > k, solve X * L11^T = A21 -> L21.
//     Tiles staged via TDM; thread t owns row t (forward substitution).
// ---------------------------------------------------------------------------
__global__ void panel_trsm(float* __restrict__ A, int k) {
    extern __shared__ float sm[];
    float* L11 = sm;               // [NB][LDT]
    float* X   = sm + NB * LDT;    // [NB][LDT]
    const int i = k + 1 + blockIdx.x;
    const int tid = threadIdx.x;   // 128 threads
    const float* dbase = A + (size_t)(k * NB) * LDA + (size_t)k * NB;
    float*       pbase = A + (size_t)(i * NB) * LDA + (size_t)k * NB;

#if HAVE_TDM
    if (tid < 32) {
        tdm_load_tile(dbase, (unsigned)(size_t)(void*)L11);
        tdm_load_tile(pbase, (unsigned)(size_t)(void*)X);
        __builtin_amdgcn_s_wait_tensorcnt(0);
    }
    __syncthreads();
#else
    for (int idx = tid; idx < NB * (NB / 4); idx += NB) {
        int r = idx >> 5, c4 = (idx & 31) << 2;
        *(float4*)(L11 + r * LDT + c4) = *(const float4*)(dbase + (size_t)r * LDA + c4);
        *(float4*)(X   + r * LDT + c4) = *(const float4*)(pbase + (size_t)r * LDA + c4);
    }
    __syncthreads();
#endif

    float* xr = X + tid * LDT;
    for (int j = 0; j < NB; ++j) {
        float sum = xr[j];
        const float* l11j = L11 + j * LDT;
        for (int p = 0; p < j; ++p) sum -= xr[p] * l11j[p];
        xr[j] = sum / l11j[j];
    }
    __syncthreads();

    for (int idx = tid; idx < NB * (NB / 4); idx += NB) {
        int r = idx >> 5, c4 = (idx & 31) << 2;
        *(float4*)(pbase + (size_t)r * LDA + c4) = *(float4*)(X + r * LDT + c4);
    }
}

// ---------------------------------------------------------------------------
// 2c) SYRK trailing update with WMMA fp32: A[ib][jb] -= P_i * P_j^T.
//     256 threads = 8 waves in a 2x4 wave grid; each wave computes a 64x32
//     sub-tile (4 M-tiles x 2 N-tiles): 6 DS frag loads per 8 WMMAs.
//     Panels staged into padded LDS via TDM.
// ---------------------------------------------------------------------------
__global__ void syrk_update(float* __restrict__ A, int k) {
    extern __shared__ float sm[];
    float* Pi = sm;               // [NB][LDT]
    float* Pj = sm + NB * LDT;    // [NB][LDT]
    const int jb = k + 1 + blockIdx.y;
    const int ib = k + 1 + blockIdx.x;
    if (ib < jb) return;          // workgroup-uniform: EXEC stays all-ones for WMMA

    const int tid = threadIdx.x;  // 256 threads
    const float* gi = A + (size_t)(ib * NB) * LDA + (size_t)k * NB;
    const float* gj = A + (size_t)(jb * NB) * LDA + (size_t)k * NB;

#if HAVE_TDM
    if (tid < 32) {
        tdm_load_tile(gi, (unsigned)(size_t)(void*)Pi);
        tdm_load_tile(gj, (unsigned)(size_t)(void*)Pj);
        __builtin_amdgcn_s_wait_tensorcnt(0);
    }
    __syncthreads();
#else
    for (int idx = tid; idx < NB * (NB / 4); idx += 256) {
        int r = idx >> 5, c4 = (idx & 31) << 2;
        *(float4*)(Pi + r * LDT + c4) = *(const float4*)(gi + (size_t)r * LDA + c4);
        *(float4*)(Pj + r * LDT + c4) = *(const float4*)(gj + (size_t)r * LDA + c4);
    }
    __syncthreads();
#endif

    const int wave = tid >> 5;
    const int lane = tid & 31;
    const int lr   = lane & 15;        // M (A-frag) / N (B-frag) within 16-tile
    const int hi   = lane >> 4;        // lane-half selects K pair
    const int wm   = wave >> 2;        // 0..1 : wave row   -> rows  [wm*64, wm*64+64)
    const int wn   = wave & 3;         // 0..3 : wave col   -> cols  [wn*32, wn*32+32)

    v8f acc[4][2];
#pragma unroll
    for (int mi = 0; mi < 4; ++mi)
#pragma unroll
        for (int ni = 0; ni < 2; ++ni)
            acc[mi][ni] = (v8f){0.f, 0.f, 0.f, 0.f, 0.f, 0.f, 0.f, 0.f};

    for (int kk = 0; kk < NB; kk += 4) {
        const int kb = kk + (hi << 1);
        v2f a[4], b[2];
#pragma unroll
        for (int mi = 0; mi < 4; ++mi) {
            // A-fragment (16x4 f32): lanes 0-15 -> K=kk,kk+1 ; lanes 16-31 -> K=kk+2,kk+3
            a[mi].x = Pi[(wm * 64 + mi * 16 + lr) * LDT + kb];
            a[mi].y = Pi[(wm * 64 + mi * 16 + lr) * LDT + kb + 1];
        }
#pragma unroll
        for (int ni = 0; ni < 2; ++ni) {
            // B-fragment (4x16): B[k][n] = P_j[n][k], mirrored half-lane layout
            b[ni].x = Pj[(wn * 32 + ni * 16 + lr) * LDT + kb];
            b[ni].y = Pj[(wn * 32 + ni * 16 + lr) * LDT + kb + 1];
        }
#pragma unroll
        for (int mi = 0; mi < 4; ++mi)
#pragma unroll
            for (int ni = 0; ni < 2; ++ni)
                acc[mi][ni] = __builtin_amdgcn_wmma_f32_16x16x4_f32(
                    /*neg_a=*/false, a[mi], /*neg_b=*/false, b[ni],
                    /*c_mod=*/(short)0, acc[mi][ni],
                    /*reuse_a=*/false, /*reuse_b=*/false);
    }

    // C/D layout: VGPR r -> M = base + r + 8*hi, N = base + lr
    float* gout = A + (size_t)(ib * NB) * LDA + (size_t)jb * NB;
#pragma unroll
    for (int mi = 0; mi < 4; ++mi) {
        const int row0 = wm * 64 + mi * 16 + (hi << 3);
#pragma unroll
        for (int ni = 0; ni < 2; ++ni) {
            const int col = wn * 32 + ni * 16 + lr;
#pragma unroll
            for (int r = 0; r < 8; ++r) {
                gout[(size_t)(row0 + r) * LDA + col] -= acc[mi][ni][r];
            }
        }
    }
}

// ---------------------------------------------------------------------------
// 3a) Forward-solve partial GEMV: chunk c of block b:
//     sbuf[c*NB + r] = sum_p L[bNB+r][cNB+p] * z[cNB+p]   (deterministic slots)
// ---------------------------------------------------------------------------
__global__ void fsolve_gemv(const float* __restrict__ A,
                            const float* __restrict__ z,
                            float* __restrict__ sbuf, int b) {
    extern __shared__ float sm[];
    float* tile = sm;               // [NB][LDT]
    float* zs   = sm + NB * LDT;    // [NB]
    const int c = blockIdx.x;       // chunk < b
    const int tid = threadIdx.x;    // 128 threads
    const float* g = A + (size_t)(b * NB) * LDA + (size_t)c * NB;

    for (int idx = tid; idx < NB * (NB / 4); idx += NB) {
        int r = idx >> 5, c4 = (idx & 31) << 2;
        *(float4*)(tile + r * LDT + c4) = *(const float4*)(g + (size_t)r * LDA + c4);
    }
    zs[tid] = z[c * NB + tid];
    __syncthreads();

    float sum = 0.f;
    for (int p = 0; p < NB; ++p) sum += tile[tid * LDT + p] * zs[p];
    sbuf[c * NB + tid] = sum;
}

// ---------------------------------------------------------------------------
// 3b) Solve diagonal block b: z_b = L_bb^{-1} (y_b - sum_c partial_c)
// ---------------------------------------------------------------------------
__global__ void fsolve_block(const float* __restrict__ A,
                             const float* __restrict__ y,
                             float* __restrict__ z,
                             const float* __restrict__ sbuf, int b) {
    extern __shared__ float sm[];
    float* Lbb = sm;                 // [NB][LDT]
    float* rhs = sm + NB * LDT;      // [NB]
    float* zl  = rhs + NB;           // [NB]
    const int tid = threadIdx.x;     // 128 threads
    const float* g = A + (size_t)(b * NB) * LDA + (size_t)b * NB;

    for (int idx = tid; idx < NB * (NB / 4); idx += NB) {
        int r = idx >> 5, c4 = (idx & 31) << 2;
        *(float4*)(Lbb + r * LDT + c4) = *(const float4*)(g + (size_t)r * LDA + c4);
    }
    float s = 0.f;
    for (int c = 0; c < b; ++c) s += sbuf[c * NB + tid];
    rhs[tid] = y[b * NB + tid] - s;
    __syncthreads();

    for (int j = 0; j < NB; ++j) {
        if (tid == j) zl[j] = rhs[j] / Lbb[j * LDT + j];
        __syncthreads();
        if (tid > j) rhs[tid] -= Lbb[tid * LDT + j] * zl[j];
        __syncthreads();
    }
    z[b * NB + tid] = zl[tid];
}

// ---------------------------------------------------------------------------
// 4) out = -0.5*||z||^2 - sum_i log(L_ii)     (since logdet = 2*sum log L_ii)
// ---------------------------------------------------------------------------
__global__ void finalize(const float* __restrict__ A,
                         const float* __restrict__ z,
                         float* __restrict__ out) {
    __shared__ float rq[256];
    __shared__ float rl[256];
    const int tid = threadIdx.x;
    float q = 0.f, ld = 0.f;
    for (int i = tid; i < TPTS; i += 256) {
        const float zi = z[i];
        q += zi * zi;
        ld += logf(A[(size_t)i * LDA + i]);
    }
    rq[tid] = q; rl[tid] = ld;
    __syncthreads();
    for (int s = 128; s > 0; s >>= 1) {
        if (tid < s) { rq[tid] += rq[tid + s]; rl[tid] += rl[tid + s]; }
        __syncthreads();
    }
    if (tid == 0) out[0] = -0.5f * rq[0] - rl[0];
}

// ---------------------------------------------------------------------------
// Host launcher (graph-capture safe: only kernel launches on `stream`)
// ---------------------------------------------------------------------------
extern "C" void kernel_launch(void* const* d_in, const int* in_sizes, int n_in,
                              void* d_out, int out_size, void* d_ws, size_t ws_size,
                              hipStream_t stream) {
    const float* y        = (const float*)d_in[0];
    const float* sigma_sq = (const float*)d_in[1];
    const float* lscale   = (const float*)d_in[2];
    const float* variance = (const float*)d_in[3];
    float* out = (float*)d_out;

    // Workspace layout: A (8192x8192 fp32 = 256MB) | z (8192) | sbuf (8192)
    float* A    = (float*)d_ws;
    float* z    = A + (size_t)TPTS * TPTS;
    float* sbuf = z + TPTS;

    (void)in_sizes; (void)n_in; (void)out_size; (void)ws_size;

    build_A<<<dim3(TPTS / 256, TPTS), 256, 0, stream>>>(A, sigma_sq, lscale, variance);

    const size_t lds1 = (size_t)NB * LDT * sizeof(float);          // one padded tile
    const size_t lds2 = 2 * lds1;                                  // two padded tiles

    for (int k = 0; k < NBLK; ++k) {
        panel_factor<<<1, NB, lds1, stream>>>(A, k);
        const int nrem = NBLK - 1 - k;
        if (nrem > 0) {
            panel_trsm<<<nrem, NB, lds2, stream>>>(A, k);
            syrk_update<<<dim3(nrem, nrem), 256, lds2, stream>>>(A, k);
        }
    }

    const size_t lds_g = lds1 + NB * sizeof(float);
    const size_t lds_s = lds1 + 2 * NB * sizeof(float);
    for (int b = 0; b < NBLK; ++b) {
        if (b > 0) fsolve_gemv<<<b, NB, lds_g, stream>>>(A, z, sbuf, b);
        fsolve_block<<<1, NB, lds_s, stream>>>(A, y, z, sbuf, b);
    }

    finalize<<<1, 256, 0, stream>>>(A, z, out);
}